// Model_33981781246390
// MI455X (gfx1250) — compile-verified
//
#include <hip/hip_runtime.h>
#include <math.h>

typedef __attribute__((ext_vector_type(2))) float v2f;
typedef __attribute__((ext_vector_type(8))) float v8f;

#define UCNT 80000
#define ICNT 40000
#define NCNT 120000
#define DDIM 64
#define NNZC 2880000

// ---------------------------------------------------------------------------
// Dense projection + bias + row L2-normalize using fp32 WMMA (16x16x4).
// One wave computes a 16-row x 64-col output tile. K stepped by 4.
//
// A fragment (16x4 fp32, 2 VGPRs): lanes 0-15 -> rows 0-15 holding K=0,1;
//                                  lanes 16-31 -> rows 0-15 holding K=2,3.
// B fragment (4x16 fp32, 2 VGPRs): mirrored (lanes<16: K=0,1; lanes>=16: K=2,3),
//                                  lane&15 = column.
// C/D (16x16 fp32, 8 VGPRs): VGPR r = row r (lanes<16) / row r+8 (lanes>=16),
//                            lane&15 = column.
// ---------------------------------------------------------------------------
__device__ inline v8f wmma_f32_k4(v2f a, v2f b, v8f c) {
    return __builtin_amdgcn_wmma_f32_16x16x4_f32(
        /*neg_a=*/false, a, /*neg_b=*/false, b,
        /*c_mod=*/(short)0, c, /*reuse_a=*/false, /*reuse_b=*/false);
}

template <int K>
__global__ void proj_norm_kernel(const float* __restrict__ E,   // [I, K]
                                 const float* __restrict__ W,   // [K, 64]
                                 const float* __restrict__ bias,// [64]
                                 float* __restrict__ F,         // [I, 64] normalized
                                 int I) {
    const int lane  = threadIdx.x & 31;
    const int wave  = threadIdx.x >> 5;
    const int tile  = blockIdx.x * (blockDim.x >> 5) + wave;
    const int m0    = tile * 16;
    if (m0 >= I) return;                      // wave-uniform guard (EXEC all 1s)

    const int nIdx   = lane & 15;
    const int half   = lane >> 4;             // 0 or 1
    const int kShift = half * 2;              // K sub-offset per half-wave

    v8f acc[4] = {};                          // 4 n-tiles of 16 cols

    const float* aBase = E + (size_t)(m0 + nIdx) * K + kShift;

    for (int k = 0; k < K; k += 4) {
        v2f a;
        a.x = aBase[k];
        a.y = aBase[k + 1];
#pragma unroll
        for (int t = 0; t < 4; ++t) {
            v2f b;
            b.x = W[(size_t)(k + kShift) * DDIM + t * 16 + nIdx];
            b.y = W[(size_t)(k + kShift + 1) * DDIM + t * 16 + nIdx];
            acc[t] = wmma_f32_k4(a, b, acc[t]);
        }
    }

    // bias add (same column value for all 8 rows held in this lane)
#pragma unroll
    for (int t = 0; t < 4; ++t) {
        float bv = bias[t * 16 + nIdx];
#pragma unroll
        for (int r = 0; r < 8; ++r) acc[t][r] += bv;
    }

    // row-wise L2 norm: each row is spread over 16 lanes (one half-wave);
    // xor-reduce with masks 1,2,4,8 stays within that 16-lane group.
#pragma unroll
    for (int r = 0; r < 8; ++r) {
        float s = acc[0][r] * acc[0][r] + acc[1][r] * acc[1][r] +
                  acc[2][r] * acc[2][r] + acc[3][r] * acc[3][r];
        s += __shfl_xor(s, 1);
        s += __shfl_xor(s, 2);
        s += __shfl_xor(s, 4);
        s += __shfl_xor(s, 8);
        float invn = 1.0f / fmaxf(sqrtf(s), 1e-12f);
        const int row = m0 + half * 8 + r;
#pragma unroll
        for (int t = 0; t < 4; ++t) {
            F[(size_t)row * DDIM + t * 16 + nIdx] = acc[t][r] * invn;
        }
    }
}

// ---------------------------------------------------------------------------
// COO SPMM: y[row] += val * x[col], x = concat(xu [U rows], xi [rows U..N)).
// 16 threads per edge, float4 per thread -> coalesced 256B gather + 4 atomics.
// ---------------------------------------------------------------------------
__global__ void spmm_kernel(const int* __restrict__ rows,
                            const int* __restrict__ cols,
                            const float* __restrict__ vals,
                            const float* __restrict__ xu,
                            const float* __restrict__ xi,
                            float* __restrict__ y,
                            int nnz) {
    long long tid = (long long)blockIdx.x * blockDim.x + threadIdx.x;
    int e = (int)(tid >> 4);
    if (e >= nnz) return;
    int sub = (int)(tid & 15);

    int r = rows[e];
    int c = cols[e];
    float v = vals[e];

    const float* xp = (c < UCNT) ? (xu + (size_t)c * DDIM)
                                 : (xi + (size_t)(c - UCNT) * DDIM);
    float4 xv = *(const float4*)(xp + sub * 4);
    float* yp = y + (size_t)r * DDIM + sub * 4;
    atomicAdd(yp + 0, v * xv.x);
    atomicAdd(yp + 1, v * xv.y);
    atomicAdd(yp + 2, v * xv.z);
    atomicAdd(yp + 3, v * xv.w);
}

// ---------------------------------------------------------------------------
// Elementwise: softmax(modal_weight) blend; writes modal and final = 1.5*modal
// (final = modal + spmm(adj,modal) + 0.5*modal; spmm accumulates afterwards).
// ---------------------------------------------------------------------------
__global__ void modal_kernel(const float* __restrict__ base,
                             const float* __restrict__ imgp,
                             const float* __restrict__ txtp,
                             const float* __restrict__ mw,
                             float* __restrict__ modal,
                             float* __restrict__ final_out,
                             int n4) {
    int i = blockIdx.x * blockDim.x + threadIdx.x;
    if (i >= n4) return;
    float m0 = mw[0], m1 = mw[1];
    float mx = fmaxf(m0, m1);
    float e0 = __expf(m0 - mx), e1 = __expf(m1 - mx);
    float inv = 1.0f / (e0 + e1);
    float w0 = e0 * inv, w1 = e1 * inv;

    float4 b  = ((const float4*)base)[i];
    float4 ip = ((const float4*)imgp)[i];
    float4 tp = ((const float4*)txtp)[i];
    float4 m;
    m.x = w0 * (b.x + 0.2f * ip.x) + w1 * (b.x + 0.2f * tp.x);
    m.y = w0 * (b.y + 0.2f * ip.y) + w1 * (b.y + 0.2f * tp.y);
    m.z = w0 * (b.z + 0.2f * ip.z) + w1 * (b.z + 0.2f * tp.z);
    m.w = w0 * (b.w + 0.2f * ip.w) + w1 * (b.w + 0.2f * tp.w);
    ((float4*)modal)[i] = m;
    float4 f;
    f.x = 1.5f * m.x; f.y = 1.5f * m.y; f.z = 1.5f * m.z; f.w = 1.5f * m.w;
    ((float4*)final_out)[i] = f;
}

__global__ void zero_kernel(float4* __restrict__ p, int n4) {
    int i = blockIdx.x * blockDim.x + threadIdx.x;
    if (i < n4) p[i] = make_float4(0.f, 0.f, 0.f, 0.f);
}

extern "C" void kernel_launch(void* const* d_in, const int* in_sizes, int n_in,
                              void* d_out, int out_size, void* d_ws, size_t ws_size,
                              hipStream_t stream) {
    const float* u_embs   = (const float*)d_in[0];
    const float* i_embs   = (const float*)d_in[1];
    const float* img_emb  = (const float*)d_in[2];
    const float* txt_emb  = (const float*)d_in[3];
    const float* image_W  = (const float*)d_in[4];
    const float* image_b  = (const float*)d_in[5];
    const float* text_W   = (const float*)d_in[6];
    const float* text_b   = (const float*)d_in[7];
    const float* modal_w  = (const float*)d_in[8];
    const float* adj_val  = (const float*)d_in[9];
    const float* img_val  = (const float*)d_in[10];
    const float* txt_val  = (const float*)d_in[11];
    const int*   adj_row  = (const int*)d_in[12];
    const int*   adj_col  = (const int*)d_in[13];
    const int*   img_row  = (const int*)d_in[14];
    const int*   img_col  = (const int*)d_in[15];
    const int*   txt_row  = (const int*)d_in[16];
    const int*   txt_col  = (const int*)d_in[17];

    float* ws    = (float*)d_ws;
    float* imgf  = ws;                                   // [I,64] normalized image feats
    float* txtf  = imgf + (size_t)ICNT * DDIM;           // [I,64] normalized text feats
    float* base  = txtf + (size_t)ICNT * DDIM;           // [N,64] spmm(adj, [u;i])
    float* modal = base + (size_t)NCNT * DDIM;           // [N,64]

    float* out    = (float*)d_out;
    float* final_o = out;                                // [N,64]
    float* imgp    = out + (size_t)NCNT * DDIM;          // [N,64]
    float* txtp    = out + (size_t)2 * NCNT * DDIM;      // [N,64]

    const int n4N = NCNT * DDIM / 4;                     // 1,920,000 float4s
    const int zb  = (n4N + 255) / 256;
    zero_kernel<<<zb, 256, 0, stream>>>((float4*)imgp, n4N);
    zero_kernel<<<zb, 256, 0, stream>>>((float4*)txtp, n4N);
    zero_kernel<<<zb, 256, 0, stream>>>((float4*)base, n4N);

    // 8 waves/block, 16 rows/wave -> 128 rows/block; 40000 rows -> 313 blocks
    const int pb = (ICNT / 16 + 7) / 8;
    proj_norm_kernel<1024><<<pb, 256, 0, stream>>>(img_emb, image_W, image_b, imgf, ICNT);
    proj_norm_kernel<384><<<pb, 256, 0, stream>>>(txt_emb, text_W, text_b, txtf, ICNT);

    const long long st = (long long)NNZC * 16;
    const int sb = (int)((st + 255) / 256);
    spmm_kernel<<<sb, 256, 0, stream>>>(img_row, img_col, img_val, u_embs, imgf, imgp, NNZC);
    spmm_kernel<<<sb, 256, 0, stream>>>(txt_row, txt_col, txt_val, u_embs, txtf, txtp, NNZC);
    spmm_kernel<<<sb, 256, 0, stream>>>(adj_row, adj_col, adj_val, u_embs, i_embs, base, NNZC);

    modal_kernel<<<zb, 256, 0, stream>>>(base, imgp, txtp, modal_w, modal, final_o, n4N);

    // final += spmm(adj, modal); modal is contiguous [N,64] so split at U rows
    spmm_kernel<<<sb, 256, 0, stream>>>(adj_row, adj_col, adj_val,
                                        modal, modal + (size_t)UCNT * DDIM, final_o, NNZC);
}